// SpGAT_56341380988952
// MI455X (gfx1250) — compile-verified
//
#include <hip/hip_runtime.h>
#include <math.h>

#define N_NODES 50000
#define R_REL   500
#define E_EDGES 400000
#define EN_EDGES 40000
#define EA_EDGES (E_EDGES + EN_EDGES)
#define ALPHA_  0.2f
#define EPS_    1e-12f

typedef float v2f __attribute__((ext_vector_type(2)));
typedef float v8f __attribute__((ext_vector_type(8)));

__device__ __forceinline__ float warp_sum(float v) {
#pragma unroll
  for (int m = 16; m > 0; m >>= 1) v += __shfl_xor(v, m, 32);
  return v;
}
__device__ __forceinline__ float eluf(float x) { return x > 0.f ? x : expm1f(x); }
__device__ __forceinline__ void atomAddF(float* p, float v) { unsafeAtomicAdd(p, v); }

// ---------------------------------------------------------------------------
// Generic fp32 WMMA GEMM: C[M x Ncols] = A[M x K] * B + bias   (or C += ...)
// TB=0: B is [K x Ncols] row-major.  TB=1: B is [Ncols x K] row-major.
// One wave computes a 16x64 output tile via 4x V_WMMA_F32_16X16X4_F32 per K-step.
// grid.x covers ceil(M/16)/8 row strips (8 waves/block), grid.y = Ncols/64.
// K must be a multiple of 4. Branch-free inner loop (row guard via mask).
// ---------------------------------------------------------------------------
template <int TB>
__global__ __launch_bounds__(256) void gemm_wmma_f32(
    const float* __restrict__ A, int lda,
    const float* __restrict__ B, int ldb,
    float* __restrict__ C, int ldc,
    const float* __restrict__ bias,
    int M, int K, int accumulate)
{
  const int lane = threadIdx.x & 31;
  const int wv   = threadIdx.x >> 5;
  const int l16  = lane & 15;
  const int hi   = lane >> 4;
  const int mt   = blockIdx.x * 8 + wv;
  if (mt * 16 >= M) return;
  const int row0 = mt * 16;
  const int col0 = blockIdx.y * 64;

  v8f c0, c1, c2, c3;
  if (accumulate) {
#pragma unroll
    for (int v = 0; v < 8; ++v) {
      const int r = row0 + v + hi * 8;
      const size_t rr = (size_t)((r < M) ? r : (M - 1)) * ldc;
      const float m = (r < M) ? 1.f : 0.f;
      c0[v] = C[rr + col0 + l16]      * m;
      c1[v] = C[rr + col0 + 16 + l16] * m;
      c2[v] = C[rr + col0 + 32 + l16] * m;
      c3[v] = C[rr + col0 + 48 + l16] * m;
    }
  } else {
    const float b0v = bias ? bias[col0 + l16]      : 0.f;
    const float b1v = bias ? bias[col0 + 16 + l16] : 0.f;
    const float b2v = bias ? bias[col0 + 32 + l16] : 0.f;
    const float b3v = bias ? bias[col0 + 48 + l16] : 0.f;
#pragma unroll
    for (int v = 0; v < 8; ++v) { c0[v] = b0v; c1[v] = b1v; c2[v] = b2v; c3[v] = b3v; }
  }

  // A fragment: lane row = row0 + l16 (clamped, masked), K pair = k + hi*2
  int rA = row0 + l16;
  float amask = 1.f;
  if (rA >= M) { rA = M - 1; amask = 0.f; }
  const float* Ap = A + (size_t)rA * lda + hi * 2;

  // B fragment pointers per 16-col group
  const float *Bp0, *Bp1, *Bp2, *Bp3;
  if (TB) {
    Bp0 = B + (size_t)(col0 + l16)      * ldb + hi * 2;
    Bp1 = B + (size_t)(col0 + 16 + l16) * ldb + hi * 2;
    Bp2 = B + (size_t)(col0 + 32 + l16) * ldb + hi * 2;
    Bp3 = B + (size_t)(col0 + 48 + l16) * ldb + hi * 2;
  } else {
    Bp0 = B + (size_t)(hi * 2) * ldb + col0 + l16;
    Bp1 = Bp0 + 16; Bp2 = Bp0 + 32; Bp3 = Bp0 + 48;
  }

#pragma unroll 2
  for (int k = 0; k < K; k += 4) {
    v2f a = *(const v2f*)Ap;
    Ap += 4;
    a.x *= amask; a.y *= amask;
    v2f b0, b1, b2, b3;
    if (TB) {
      b0 = *(const v2f*)Bp0; Bp0 += 4;
      b1 = *(const v2f*)Bp1; Bp1 += 4;
      b2 = *(const v2f*)Bp2; Bp2 += 4;
      b3 = *(const v2f*)Bp3; Bp3 += 4;
    } else {
      b0.x = Bp0[0]; b0.y = Bp0[ldb];
      b1.x = Bp1[0]; b1.y = Bp1[ldb];
      b2.x = Bp2[0]; b2.y = Bp2[ldb];
      b3.x = Bp3[0]; b3.y = Bp3[ldb];
      Bp0 += 4 * (size_t)ldb; Bp1 += 4 * (size_t)ldb;
      Bp2 += 4 * (size_t)ldb; Bp3 += 4 * (size_t)ldb;
    }
    c0 = __builtin_amdgcn_wmma_f32_16x16x4_f32(false, a, false, b0, (short)0, c0, false, false);
    c1 = __builtin_amdgcn_wmma_f32_16x16x4_f32(false, a, false, b1, (short)0, c1, false, false);
    c2 = __builtin_amdgcn_wmma_f32_16x16x4_f32(false, a, false, b2, (short)0, c2, false, false);
    c3 = __builtin_amdgcn_wmma_f32_16x16x4_f32(false, a, false, b3, (short)0, c3, false, false);
  }

#pragma unroll
  for (int v = 0; v < 8; ++v) {
    const int r = row0 + v + hi * 8;
    if (r < M) {
      float* Cp = C + (size_t)r * ldc + col0 + l16;
      Cp[0]  = c0[v];
      Cp[16] = c1[v];
      Cp[32] = c2[v];
      Cp[48] = c3[v];
    }
  }
}

// ---------------- weight packing -------------------------------------------
// W1[k][j] (128x256): j block b=j>>6 -> {h0 src, h0 tgt, h1 src, h1 tgt}, c=j&63
__global__ void pack_w1(const float* __restrict__ a0, const float* __restrict__ a1,
                        float* __restrict__ W1) {
  int i = blockIdx.x * 256 + threadIdx.x;
  if (i >= 128 * 256) return;
  int k = i >> 8, j = i & 255;
  int b = j >> 6, c = j & 63;
  const float* a = (b >> 1) ? a1 : a0;   // a_h: (64 x 320) row-major
  int st = b & 1;
  W1[i] = a[c * 320 + st * 128 + k];
}
// W2[k][j] (128x256): cols 0:128 = a_out_w[:, :128]^T, cols 128:256 = a_out_w[:,128:256]^T
__global__ void pack_w2(const float* __restrict__ aw, float* __restrict__ W2) {
  int i = blockIdx.x * 256 + threadIdx.x;
  if (i >= 128 * 256) return;
  int k = i >> 8, j = i & 255;
  W2[i] = (j < 128) ? aw[j * 384 + k] : aw[(j - 128) * 384 + 128 + k];
}
// Block-diagonal a_rel^T (128x128): Wbd[h*64+k][h*64+c] = a_h[c][256+k]
__global__ void pack_wbd(const float* __restrict__ a0, const float* __restrict__ a1,
                         float* __restrict__ Wbd) {
  int i = blockIdx.x * 256 + threadIdx.x;
  if (i >= 128 * 128) return;
  int j = i >> 7, col = i & 127;
  int hj = j >> 6, k = j & 63;
  int hc = col >> 6, c = col & 63;
  const float* a = hj ? a1 : a0;
  Wbd[i] = (hj == hc) ? a[c * 320 + 256 + k] : 0.f;
}
// wrel[h][k] = sum_c a2_h[c] * a_rel_h[c][k]
__global__ void make_wrel(const float* __restrict__ a0, const float* __restrict__ a1,
                          const float* __restrict__ a20, const float* __restrict__ a21,
                          float* __restrict__ wrel) {
  int i = threadIdx.x;
  if (i >= 128) return;
  int h = i >> 6, k = i & 63;
  const float* a  = h ? a1 : a0;
  const float* a2 = h ? a21 : a20;
  float s = 0.f;
  for (int c = 0; c < 64; ++c) s += a2[c] * a[c * 320 + 256 + k];
  wrel[i] = s;
}

// ---------------- per-node dot tables --------------------------------------
__global__ void calc_ps1(const float* __restrict__ s1t1, const float* __restrict__ a20,
                         const float* __restrict__ a21, float* __restrict__ ps1) {
  int i = blockIdx.x * 256 + threadIdx.x;
  if (i >= N_NODES * 4) return;
  int n = i >> 2, j = i & 3;
  const float* a2 = (j >> 1) ? a21 : a20;
  const float* row = s1t1 + (size_t)n * 256 + j * 64;
  float s = 0.f;
  for (int c = 0; c < 64; ++c) s += row[c] * a2[c];
  ps1[i] = s;
}
__global__ void calc_ps2(const float* __restrict__ s2t2, const float* __restrict__ a2o,
                         float* __restrict__ ps2) {
  int i = blockIdx.x * 256 + threadIdx.x;
  if (i >= N_NODES * 2) return;
  int n = i >> 1, j = i & 1;
  const float* row = s2t2 + (size_t)n * 256 + j * 128;
  float s = 0.f;
  for (int c = 0; c < 128; ++c) s += row[c] * a2o[c];
  ps2[i] = s;
}
__global__ void calc_pr2(const float* __restrict__ r2, const float* __restrict__ a2o,
                         float* __restrict__ pr2) {
  int t = blockIdx.x * 256 + threadIdx.x;
  if (t >= R_REL) return;
  float s = 0.f;
  for (int c = 0; c < 128; ++c) s += r2[(size_t)t * 128 + c] * a2o[c];
  pr2[t] = s;
}

// ---------------- layer-1 fused edge pass ----------------------------------
__global__ __launch_bounds__(256) void edge_pass1(
    const int* __restrict__ el, const int* __restrict__ nh, const int* __restrict__ nht,
    const float* __restrict__ edge_embed, const float* __restrict__ rel_embed,
    const float* __restrict__ s1t1, const float* __restrict__ ps1,
    const float* __restrict__ wrel,
    float* __restrict__ hacc, float* __restrict__ accR, float* __restrict__ rowsum)
{
  const int lane = threadIdx.x & 31;
  const int e = blockIdx.x * 8 + (threadIdx.x >> 5);
  if (e >= EA_EDGES) return;
  const int h = blockIdx.y & 1;
  const int d = blockIdx.y >> 1;

  int a0, a1;
  float emx, emy;
  if (e < E_EDGES) {
    a0 = el[e]; a1 = el[E_EDGES + e];
    const float* p = edge_embed + (size_t)e * 64 + lane * 2;
    emx = p[0]; emy = p[1];
  } else {
    int en = e - E_EDGES;
    a0 = nh[en]; a1 = nh[EN_EDGES + en];
    int t0 = nht[en * 2], t1 = nht[en * 2 + 1];
    const float* p0 = rel_embed + (size_t)t0 * 64 + lane * 2;
    const float* p1 = rel_embed + (size_t)t1 * 64 + lane * 2;
    emx = p0[0] + p1[0]; emy = p0[1] + p1[1];
  }
  const int src = d ? a1 : a0;
  const int tgt = d ? a0 : a1;

  const float* wr = wrel + h * 64 + lane * 2;
  float pe = warp_sum(emx * wr[0] + emy * wr[1]);
  float p = ps1[src * 4 + 2 * h] + ps1[tgt * 4 + 2 * h + 1] + pe;
  float lr = p > 0.f ? p : ALPHA_ * p;
  float w = __expf(-lr);

  if (lane == 0) atomAddF(&rowsum[(size_t)(d * 2 + h) * N_NODES + src], w);

  const float* tv = s1t1 + (size_t)tgt * 256 + (2 * h + 1) * 64 + lane * 2;
  size_t base = ((size_t)(d * N_NODES + src)) * 128 + h * 64 + lane * 2;
  atomAddF(hacc + base,     w * tv[0]);
  atomAddF(hacc + base + 1, w * tv[1]);
  atomAddF(accR + base,     w * emx);
  atomAddF(accR + base + 1, w * emy);
}

// ---------------- layer-2 fused edge pass ----------------------------------
__global__ __launch_bounds__(256) void edge_pass2(
    const int* __restrict__ el, const int* __restrict__ nh,
    const int* __restrict__ et, const int* __restrict__ nht,
    const float* __restrict__ s2t2, const float* __restrict__ ps2,
    const float* __restrict__ r2, const float* __restrict__ pr2,
    float* __restrict__ hacc2, float* __restrict__ rowsum2)
{
  const int lane = threadIdx.x & 31;
  const int e = blockIdx.x * 8 + (threadIdx.x >> 5);
  if (e >= EA_EDGES) return;
  const int d = blockIdx.y;

  int a0, a1;
  float r0, r1, r2v, r3, pr;
  if (e < E_EDGES) {
    a0 = el[e]; a1 = el[E_EDGES + e];
    int t = et[e];
    const float* rp = r2 + (size_t)t * 128 + lane * 4;
    r0 = rp[0]; r1 = rp[1]; r2v = rp[2]; r3 = rp[3];
    pr = pr2[t];
  } else {
    int en = e - E_EDGES;
    a0 = nh[en]; a1 = nh[EN_EDGES + en];
    int t0 = nht[en * 2], t1 = nht[en * 2 + 1];
    const float* p0 = r2 + (size_t)t0 * 128 + lane * 4;
    const float* p1 = r2 + (size_t)t1 * 128 + lane * 4;
    r0 = p0[0] + p1[0]; r1 = p0[1] + p1[1];
    r2v = p0[2] + p1[2]; r3 = p0[3] + p1[3];
    pr = pr2[t0] + pr2[t1];
  }
  const int src = d ? a1 : a0;
  const int tgt = d ? a0 : a1;

  float p = ps2[src * 2] + ps2[tgt * 2 + 1] + pr;
  float lr = p > 0.f ? p : ALPHA_ * p;
  float w = __expf(-lr);

  if (lane == 0) atomAddF(&rowsum2[(size_t)d * N_NODES + src], w);

  const float* tv = s2t2 + (size_t)tgt * 256 + 128 + lane * 4;
  float* hp = hacc2 + ((size_t)(d * N_NODES + src)) * 128 + lane * 4;
  atomAddF(hp,     w * (tv[0] + r0));
  atomAddF(hp + 1, w * (tv[1] + r1));
  atomAddF(hp + 2, w * (tv[2] + r2v));
  atomAddF(hp + 3, w * (tv[3] + r3));
}

// ---------------- normalize + elu (in place on hacc) -----------------------
__global__ void normalize1(float* __restrict__ hacc, const float* __restrict__ s1t1,
                           const float* __restrict__ rowsum) {
  int i = blockIdx.x * 256 + threadIdx.x;
  if (i >= 2 * N_NODES * 128) return;
  int j = i & 127;
  int dn = i >> 7;          // d*N + n
  int n = dn % N_NODES;
  int d = dn / N_NODES;
  int h = j >> 6, c = j & 63;
  float rs = rowsum[(size_t)(d * 2 + h) * N_NODES + n];
  float s = s1t1[(size_t)n * 256 + (2 * h) * 64 + c];
  float v = (rs > 0.f) ? (s + hacc[i] / rs) : 0.f;
  hacc[i] = eluf(v);
}
__global__ void normalize2(float* __restrict__ hacc, const float* __restrict__ s2t2,
                           const float* __restrict__ rowsum) {
  int i = blockIdx.x * 256 + threadIdx.x;
  if (i >= 2 * N_NODES * 128) return;
  int j = i & 127;
  int dn = i >> 7;
  int n = dn % N_NODES;
  int d = dn / N_NODES;
  float rs = rowsum[(size_t)d * N_NODES + n];
  float s = s2t2[(size_t)n * 256 + j];
  float v = (rs > 0.f) ? (s + hacc[i] / rs) : 0.f;
  hacc[i] = eluf(v);
}

// ---------------- g = segment_sum(edge_embed, edge_type) -------------------
__global__ void scatter_g(const float* __restrict__ edge_embed, const int* __restrict__ et,
                          float* __restrict__ g) {
  int i = blockIdx.x * 256 + threadIdx.x;     // e*16 + q
  if (i >= E_EDGES * 16) return;
  int e = i >> 4, q = i & 15;
  int t = et[e];
  const float* p = edge_embed + (size_t)e * 64 + q * 4;
  float* gp = g + (size_t)t * 64 + q * 4;
  atomAddF(gp,     p[0]);
  atomAddF(gp + 1, p[1]);
  atomAddF(gp + 2, p[2]);
  atomAddF(gp + 3, p[3]);
}

// ---------------- row-wise L2 normalize (128 cols, wave/row) ---------------
__global__ void l2n_rows(float* __restrict__ X, int rows) {
  const int lane = threadIdx.x & 31;
  const int r = blockIdx.x * 8 + (threadIdx.x >> 5);
  if (r >= rows) return;
  float* p = X + (size_t)r * 128 + lane * 4;
  float v0 = p[0], v1 = p[1], v2 = p[2], v3 = p[3];
  float ss = warp_sum(v0 * v0 + v1 * v1 + v2 * v2 + v3 * v3);
  float inv = 1.f / fmaxf(sqrtf(ss), EPS_);
  p[0] = v0 * inv; p[1] = v1 * inv; p[2] = v2 * inv; p[3] = v3 * inv;
}

// ---------------- merge gate + elu + l2n (wave/row) ------------------------
__global__ void merge_combine(const float* __restrict__ hi, const float* __restrict__ ho,
                              const float* __restrict__ lw, const float* __restrict__ lb,
                              float* __restrict__ out) {
  const int lane = threadIdx.x & 31;
  const int n = blockIdx.x * 8 + (threadIdx.x >> 5);
  if (n >= N_NODES) return;
  const float* ap = hi + (size_t)n * 128 + lane * 4;
  const float* bp = ho + (size_t)n * 128 + lane * 4;
  const float* w1 = lw + lane * 4;
  const float* w2 = lw + 128 + lane * 4;
  float a0 = ap[0], a1 = ap[1], a2 = ap[2], a3 = ap[3];
  float b0 = bp[0], b1 = bp[1], b2 = bp[2], b3 = bp[3];
  float part = a0 * w1[0] + a1 * w1[1] + a2 * w1[2] + a3 * w1[3]
             + b0 * w2[0] + b1 * w2[1] + b2 * w2[2] + b3 * w2[3];
  float gsum = warp_sum(part);
  float lam = 1.f / (1.f + __expf(-(gsum + lb[0])));
  float v0 = eluf(lam * a0 + (1.f - lam) * b0);
  float v1 = eluf(lam * a1 + (1.f - lam) * b1);
  float v2 = eluf(lam * a2 + (1.f - lam) * b2);
  float v3 = eluf(lam * a3 + (1.f - lam) * b3);
  float ss = warp_sum(v0 * v0 + v1 * v1 + v2 * v2 + v3 * v3);
  float inv = 1.f / fmaxf(sqrtf(ss), EPS_);
  float* op = out + (size_t)n * 128 + lane * 4;
  op[0] = v0 * inv; op[1] = v1 * inv; op[2] = v2 * inv; op[3] = v3 * inv;
}

// ===========================================================================
extern "C" void kernel_launch(void* const* d_in, const int* in_sizes, int n_in,
                              void* d_out, int out_size, void* d_ws, size_t ws_size,
                              hipStream_t stream) {
  (void)in_sizes; (void)n_in; (void)out_size; (void)ws_size;
  const float* X      = (const float*)d_in[0];
  const float* REL    = (const float*)d_in[1];
  const float* EEMB   = (const float*)d_in[2];
  const float* AH0    = (const float*)d_in[3];
  const float* A20    = (const float*)d_in[4];
  const float* AH1    = (const float*)d_in[5];
  const float* A21    = (const float*)d_in[6];
  const float* AOW    = (const float*)d_in[7];
  const float* A2O    = (const float*)d_in[8];
  const float* MIWIN  = (const float*)d_in[9];
  const float* MIBIN  = (const float*)d_in[10];
  const float* MIWOUT = (const float*)d_in[11];
  const float* MIBOUT = (const float*)d_in[12];
  const float* MILW   = (const float*)d_in[13];
  const float* MILB   = (const float*)d_in[14];
  const float* MOWIN  = (const float*)d_in[15];
  const float* MOBIN  = (const float*)d_in[16];
  const float* MOWOUT = (const float*)d_in[17];
  const float* MOBOUT = (const float*)d_in[18];
  const float* MOLW   = (const float*)d_in[19];
  const float* MOLB   = (const float*)d_in[20];
  const float* RELWR  = (const float*)d_in[21];
  const float* RELW   = (const float*)d_in[22];
  const int*   EL     = (const int*)d_in[23];
  const int*   ET     = (const int*)d_in[24];
  const int*   NH     = (const int*)d_in[25];
  const int*   NHT    = (const int*)d_in[26];

  float* OUT    = (float*)d_out;
  float* OUTREL = OUT + (size_t)N_NODES * 128;

  char* base = (char*)d_ws;
  size_t off = 0;
  auto alloc = [&](size_t bytes) -> float* {
    float* p = (float*)(base + off);
    off += (bytes + 255) & ~(size_t)255;
    return p;
  };
  float* s1t1 = alloc((size_t)N_NODES * 256 * 4);      // also merge hi/ho temps
  float* hacc = alloc((size_t)2 * N_NODES * 128 * 4);  // hacc1 then hacc2
  float* accR = alloc((size_t)2 * N_NODES * 128 * 4);  // accR then s2t2
  float* xbuf = alloc((size_t)N_NODES * 128 * 4);
  float* rowsum1 = alloc((size_t)4 * N_NODES * 4);
  float* rowsum2 = alloc((size_t)2 * N_NODES * 4);
  float* ps1 = alloc((size_t)N_NODES * 4 * 4);
  float* ps2 = alloc((size_t)N_NODES * 2 * 4);
  float* W1  = alloc(128 * 256 * 4);
  float* W2  = alloc(128 * 256 * 4);
  float* Wbd = alloc(128 * 128 * 4);
  float* wrel = alloc(128 * 4);
  float* gbuf = alloc((size_t)R_REL * 64 * 4);
  float* r2   = alloc((size_t)R_REL * 128 * 4);
  float* pr2  = alloc((size_t)R_REL * 4);

  hipMemsetAsync(hacc, 0, (size_t)2 * N_NODES * 128 * 4, stream);
  hipMemsetAsync(accR, 0, (size_t)2 * N_NODES * 128 * 4, stream);
  hipMemsetAsync(rowsum1, 0, (size_t)4 * N_NODES * 4, stream);
  hipMemsetAsync(rowsum2, 0, (size_t)2 * N_NODES * 4, stream);
  hipMemsetAsync(gbuf, 0, (size_t)R_REL * 64 * 4, stream);

  pack_w1<<<128, 256, 0, stream>>>(AH0, AH1, W1);
  pack_w2<<<128, 256, 0, stream>>>(AOW, W2);
  pack_wbd<<<64, 256, 0, stream>>>(AH0, AH1, Wbd);
  make_wrel<<<1, 128, 0, stream>>>(AH0, AH1, A20, A21, wrel);

  const int MT_N  = (N_NODES / 16 + 7) / 8;        // 391
  const int MT_2N = (2 * N_NODES / 16 + 7) / 8;    // 782
  const int MT_R  = ((R_REL + 15) / 16 + 7) / 8;   // 4

  // s1t1 = X @ W1  (N x 256)
  gemm_wmma_f32<0><<<dim3(MT_N, 4), 256, 0, stream>>>(X, 128, W1, 256, s1t1, 256,
                                                      nullptr, N_NODES, 128, 0);
  calc_ps1<<<(N_NODES * 4 + 255) / 256, 256, 0, stream>>>(s1t1, A20, A21, ps1);

  edge_pass1<<<dim3(EA_EDGES / 8, 4), 256, 0, stream>>>(EL, NH, NHT, EEMB, REL,
                                                        s1t1, ps1, wrel,
                                                        hacc, accR, rowsum1);
  // hacc1 += accR @ blockdiag(a_rel^T)
  gemm_wmma_f32<0><<<dim3(MT_2N, 2), 256, 0, stream>>>(accR, 128, Wbd, 128, hacc, 128,
                                                       nullptr, 2 * N_NODES, 128, 1);
  normalize1<<<(2 * N_NODES * 128 + 255) / 256, 256, 0, stream>>>(hacc, s1t1, rowsum1);

  // merge 1: hi/ho into s1t1 region, result -> xbuf
  gemm_wmma_f32<1><<<dim3(MT_N, 2), 256, 0, stream>>>(hacc, 128, MIWIN, 128,
                                                      s1t1, 128, MIBIN, N_NODES, 128, 0);
  gemm_wmma_f32<1><<<dim3(MT_N, 2), 256, 0, stream>>>(hacc + (size_t)N_NODES * 128, 128,
                                                      MIWOUT, 128,
                                                      s1t1 + (size_t)N_NODES * 128, 128,
                                                      MIBOUT, N_NODES, 128, 0);
  merge_combine<<<(N_NODES + 7) / 8, 256, 0, stream>>>(s1t1, s1t1 + (size_t)N_NODES * 128,
                                                       MILW, MILB, xbuf);

  // out_rel = l2n(REL @ rel_wr^T + g @ rel_W)
  scatter_g<<<(E_EDGES * 16) / 256, 256, 0, stream>>>(EEMB, ET, gbuf);
  gemm_wmma_f32<1><<<dim3(MT_R, 2), 256, 0, stream>>>(REL, 64, RELWR, 64, OUTREL, 128,
                                                      nullptr, R_REL, 64, 0);
  gemm_wmma_f32<0><<<dim3(MT_R, 2), 256, 0, stream>>>(gbuf, 64, RELW, 128, OUTREL, 128,
                                                      nullptr, R_REL, 64, 1);
  l2n_rows<<<(R_REL + 7) / 8, 256, 0, stream>>>(OUTREL, R_REL);

  // r2 = out_rel @ a_rel2^T, pr2 = r2 . a2_out
  gemm_wmma_f32<1><<<dim3(MT_R, 2), 256, 0, stream>>>(OUTREL, 128, AOW + 256, 384,
                                                      r2, 128, nullptr, R_REL, 128, 0);
  calc_pr2<<<2, 256, 0, stream>>>(r2, A2O, pr2);

  // layer 2: s2t2 (in accR region), reset hacc as hacc2
  hipMemsetAsync(hacc, 0, (size_t)2 * N_NODES * 128 * 4, stream);
  gemm_wmma_f32<0><<<dim3(MT_N, 4), 256, 0, stream>>>(xbuf, 128, W2, 256, accR, 256,
                                                      nullptr, N_NODES, 128, 0);
  calc_ps2<<<(N_NODES * 2 + 255) / 256, 256, 0, stream>>>(accR, A2O, ps2);

  edge_pass2<<<dim3(EA_EDGES / 8, 2), 256, 0, stream>>>(EL, NH, ET, NHT, accR, ps2,
                                                        r2, pr2, hacc, rowsum2);
  normalize2<<<(2 * N_NODES * 128 + 255) / 256, 256, 0, stream>>>(hacc, accR, rowsum2);

  // merge 2 -> final x into d_out
  gemm_wmma_f32<1><<<dim3(MT_N, 2), 256, 0, stream>>>(hacc, 128, MOWIN, 128,
                                                      s1t1, 128, MOBIN, N_NODES, 128, 0);
  gemm_wmma_f32<1><<<dim3(MT_N, 2), 256, 0, stream>>>(hacc + (size_t)N_NODES * 128, 128,
                                                      MOWOUT, 128,
                                                      s1t1 + (size_t)N_NODES * 128, 128,
                                                      MOBOUT, N_NODES, 128, 0);
  merge_combine<<<(N_NODES + 7) / 8, 256, 0, stream>>>(s1t1, s1t1 + (size_t)N_NODES * 128,
                                                       MOLW, MOLB, OUT);
}